// GNN_24661702214198
// MI455X (gfx1250) — compile-verified
//
#include <hip/hip_runtime.h>
#include <hip/hip_bf16.h>

// ---------------------------------------------------------------------------
// GraphSAGE (2 layers) on MI455X / gfx1250.
// Memory-bound workload (~1.7 GB DRAM -> ~75us @ 23.3 TB/s); GEMMs (19.7
// GFLOP fp32) run on V_WMMA_F32_16X16X4_F32 so precision matches the fp32
// reference. The GEMM keeps 8 independent 16x16 accumulator tiles live so
// consecutive WMMAs have no RAW hazard and issue at pipeline rate.
// ---------------------------------------------------------------------------

#define GN_N 100000
#define GN_E 640000
#define GN_D 128

typedef __attribute__((ext_vector_type(2))) float v2f;
typedef __attribute__((ext_vector_type(8))) float v8f;

// ---------------------------------------------------------------------------
// Pack a row-major 128x128 weight W (out[j] = sum_k A[k]*W[j][k]) into
// WMMA B-fragment order for V_WMMA_F32_16X16X4_F32:
//   B is (4 x 16) per step; VGPR v, lane-half h holds row K = k0 + 2h + v,
//   column N = lane&15. We store, for each (k0/4, h, j), the contiguous pair
//   (W[j][k0+2h], W[j][k0+2h+1]) so the GEMM kernel issues one b64 load.
// ---------------------------------------------------------------------------
__global__ void sage_pack_w(const float* __restrict__ W, float* __restrict__ Bpk) {
    int idx = blockIdx.x * blockDim.x + threadIdx.x;   // one thread per float2
    if (idx >= 32 * 2 * 128) return;                   // 8192 pairs
    int j  = idx & 127;
    int h  = (idx >> 7) & 1;
    int kg = idx >> 8;                                  // k0 / 4
    int k  = kg * 4 + 2 * h;
    Bpk[idx * 2 + 0] = W[j * GN_D + k];
    Bpk[idx * 2 + 1] = W[j * GN_D + k + 1];
}

// ---------------------------------------------------------------------------
// Degree: atomic count per destination, then invert (shared by both layers).
// ---------------------------------------------------------------------------
__global__ void sage_count(const int* __restrict__ dst, float* __restrict__ cnt) {
    int e = blockIdx.x * blockDim.x + threadIdx.x;
    if (e < GN_E) atomicAdd(&cnt[dst[e]], 1.0f);
}

__global__ void sage_invdeg(float* __restrict__ cnt) {
    int i = blockIdx.x * blockDim.x + threadIdx.x;
    if (i < GN_N) cnt[i] = 1.0f / fmaxf(cnt[i], 1.0f);
}

// ---------------------------------------------------------------------------
// Edge scatter: one wave32 per edge. float4 coalesced gather of the source's
// 128 features, 4x global_atomic_add_f32 into SUM[dst].
// ---------------------------------------------------------------------------
__global__ __launch_bounds__(256) void sage_scatter(
    const float* __restrict__ H, const int* __restrict__ src,
    const int* __restrict__ dst, float* __restrict__ SUM) {
    int idx  = blockIdx.x * blockDim.x + threadIdx.x;
    int e    = idx >> 5;
    int lane = idx & 31;
    if (e >= GN_E) return;
    int s = src[e];
    int d = dst[e];
    const float4 v = *(const float4*)(H + (size_t)s * GN_D + lane * 4);
    float* p = SUM + (size_t)d * GN_D + lane * 4;
    atomicAdd(p + 0, v.x);
    atomicAdd(p + 1, v.y);
    atomicAdd(p + 2, v.z);
    atomicAdd(p + 3, v.w);
}

// ---------------------------------------------------------------------------
// WMMA GEMM: OUT[16rows x 128] = act( scale(A1) @ B1 + bias [+ A2 @ B2] )
//   - one wave per 16-row block (100000 = 16 * 6250 exactly)
//   - all 8 column tiles (8 x v8f accumulators) held live; K is the outer
//     loop, so the 8 WMMAs per K-step are independent (no RAW chain) and
//     each A fragment is loaded exactly once, each B word exactly once.
//   - invdeg != nullptr applies per-row 1/deg to A1 (segment-mean fused)
//   - A2/B2 != nullptr accumulates the root-path GEMM into the same C regs
// A fragment (ISA 16x4 layout): lanes 0-15 -> K = k0,k0+1; lanes 16-31 ->
// K = k0+2,k0+3; row = rbase + (lane&15). Natural row-major float2 load.
// ---------------------------------------------------------------------------
__global__ __launch_bounds__(128) void sage_gemm(
    const float* __restrict__ A1, const float* __restrict__ invdeg,
    const float* __restrict__ B1pk, const float* __restrict__ bias,
    const float* __restrict__ A2, const float* __restrict__ B2pk,
    float* __restrict__ out, int relu_out) {
    const int lane = threadIdx.x & 31;
    const int wave = blockIdx.x * (blockDim.x >> 5) + (threadIdx.x >> 5);
    if (wave >= GN_N / 16) return;

    const int rbase = wave * 16;
    const int h     = lane >> 4;        // lane half selects K pair
    const int l15   = lane & 15;
    const int row   = rbase + l15;      // A row fed by this lane

    float s = 1.0f;
    if (invdeg) s = invdeg[row];

    const float* a1row = A1 + (size_t)row * GN_D;
    const float* a2row = A2 ? (A2 + (size_t)row * GN_D) : nullptr;

    // preload bias (C column = lane&15 for every accumulator VGPR)
    float bj[8];
#pragma unroll
    for (int jt = 0; jt < 8; ++jt) bj[jt] = bias[jt * 16 + l15];

    v8f c[8];
#pragma unroll
    for (int jt = 0; jt < 8; ++jt)
#pragma unroll
        for (int v = 0; v < 8; ++v) c[jt][v] = bj[jt];

    // aggregation path: (SUM * invdeg) @ Wl^T   (or X @ Wp^T when s==1)
#pragma unroll
    for (int k0 = 0; k0 < GN_D; k0 += 4) {
        v2f a = *(const v2f*)(a1row + k0 + 2 * h);
        a.x *= s;
        a.y *= s;
        const float* brow = B1pk + ((k0 >> 2) * 2 + h) * 256 + 2 * l15;
#pragma unroll
        for (int jt = 0; jt < 8; ++jt) {
            v2f b = *(const v2f*)(brow + jt * 32);
            c[jt] = __builtin_amdgcn_wmma_f32_16x16x4_f32(
                false, a, false, b, (short)0, c[jt], false, false);
        }
    }

    // root path: X @ Wr^T accumulated into the same C tiles
    if (a2row) {
#pragma unroll
        for (int k0 = 0; k0 < GN_D; k0 += 4) {
            v2f a = *(const v2f*)(a2row + k0 + 2 * h);
            const float* brow = B2pk + ((k0 >> 2) * 2 + h) * 256 + 2 * l15;
#pragma unroll
            for (int jt = 0; jt < 8; ++jt) {
                v2f b = *(const v2f*)(brow + jt * 32);
                c[jt] = __builtin_amdgcn_wmma_f32_16x16x4_f32(
                    false, a, false, b, (short)0, c[jt], false, false);
            }
        }
    }

    // C/D layout: VGPR v, half h -> M = v + 8h, N = lane&15
#pragma unroll
    for (int jt = 0; jt < 8; ++jt)
#pragma unroll
        for (int v = 0; v < 8; ++v) {
            float val = c[jt][v];
            if (relu_out) val = fmaxf(val, 0.0f);
            out[(size_t)(rbase + v + 8 * h) * GN_D + jt * 16 + l15] = val;
        }
}

// ---------------------------------------------------------------------------
// Host orchestration (graph-capture safe: only kernel launches + memsetAsync)
// ---------------------------------------------------------------------------
extern "C" void kernel_launch(void* const* d_in, const int* in_sizes, int n_in,
                              void* d_out, int out_size, void* d_ws, size_t ws_size,
                              hipStream_t stream) {
    (void)in_sizes; (void)n_in; (void)out_size; (void)ws_size;

    const float* x   = (const float*)d_in[0];
    const int*   ei  = (const int*)d_in[1];        // [2,E]: src = ei, dst = ei+E
    const float* W0p = (const float*)d_in[2];
    const float* b0p = (const float*)d_in[3];
    const float* W0l = (const float*)d_in[4];
    const float* b0l = (const float*)d_in[5];
    const float* W0r = (const float*)d_in[6];
    const float* W1p = (const float*)d_in[7];
    const float* b1p = (const float*)d_in[8];
    const float* W1l = (const float*)d_in[9];
    const float* b1l = (const float*)d_in[10];
    const float* W1r = (const float*)d_in[11];
    float* out = (float*)d_out;

    const size_t nd = (size_t)GN_N * GN_D;
    float* H   = (float*)d_ws;       // N*D
    float* X1  = H + nd;             // N*D  (layer-0 output)
    float* SUM = X1 + nd;            // N*D
    float* INV = SUM + nd;           // N    (degree -> 1/deg)
    float* PK  = INV + GN_N;         // 6 * 16384 packed weights
    float* P0p = PK + 0 * 16384;
    float* P0l = PK + 1 * 16384;
    float* P0r = PK + 2 * 16384;
    float* P1p = PK + 3 * 16384;
    float* P1l = PK + 4 * 16384;
    float* P1r = PK + 5 * 16384;

    const int* src = ei;
    const int* dst = ei + GN_E;

    // --- weight packing (6 x 8192 pairs) ---
    sage_pack_w<<<32, 256, 0, stream>>>(W0p, P0p);
    sage_pack_w<<<32, 256, 0, stream>>>(W0l, P0l);
    sage_pack_w<<<32, 256, 0, stream>>>(W0r, P0r);
    sage_pack_w<<<32, 256, 0, stream>>>(W1p, P1p);
    sage_pack_w<<<32, 256, 0, stream>>>(W1l, P1l);
    sage_pack_w<<<32, 256, 0, stream>>>(W1r, P1r);

    // --- degrees (graph identical for both layers) ---
    hipMemsetAsync(INV, 0, GN_N * sizeof(float), stream);
    sage_count<<<(GN_E + 255) / 256, 256, 0, stream>>>(dst, INV);
    sage_invdeg<<<(GN_N + 255) / 256, 256, 0, stream>>>(INV);

    const int gemm_blocks    = (GN_N / 16 + 3) / 4;            // 4 waves/block
    const int scatter_blocks = (GN_E * 32 + 255) / 256;

    // ===== layer 0 =====
    sage_gemm<<<gemm_blocks, 128, 0, stream>>>(x, nullptr, P0p, b0p,
                                               nullptr, nullptr, H, 1);
    hipMemsetAsync(SUM, 0, nd * sizeof(float), stream);
    sage_scatter<<<scatter_blocks, 256, 0, stream>>>(H, src, dst, SUM);
    sage_gemm<<<gemm_blocks, 128, 0, stream>>>(SUM, INV, P0l, b0l,
                                               x, P0r, X1, 1);

    // ===== layer 1 =====
    sage_gemm<<<gemm_blocks, 128, 0, stream>>>(X1, nullptr, P1p, b1p,
                                               nullptr, nullptr, H, 1);
    hipMemsetAsync(SUM, 0, nd * sizeof(float), stream);
    sage_scatter<<<scatter_blocks, 256, 0, stream>>>(H, src, dst, SUM);
    sage_gemm<<<gemm_blocks, 128, 0, stream>>>(SUM, INV, P1l, b1l,
                                               X1, P1r, out, 0);
}